// VMATDosePredictorAttention_72739566125249
// MI455X (gfx1250) — compile-verified
//
#include <hip/hip_runtime.h>
#include <hip/hip_bf16.h>
#include <math.h>

// Problem constants (fixed by the reference setup)
#define BATCH 2
#define C_CT  128
#define C_B   32
#define NCTRL 180
#define VOX   13824        // 24*24*24
#define SEG   10           // N-dimension split for occupancy / MLP
#define NSEG  (NCTRL / SEG) // 18

typedef __attribute__((ext_vector_type(2))) float v2f;
typedef __attribute__((ext_vector_type(8))) float v8f;

// ---------------------------------------------------------------------------
// Kernel 1: query projection  Q[b,v,o] = sum_c ct[b,c,v] * qw[o,c] + qb[o]
// One wave computes a 16-voxel x 32-channel tile via V_WMMA_F32_16X16X4_F32.
// ---------------------------------------------------------------------------
__global__ __launch_bounds__(256)
void qproj_wmma(const float* __restrict__ ct,   // [B, C_CT, VOX]
                const float* __restrict__ qw,   // [C_B, C_CT]
                const float* __restrict__ qb,   // [C_B]
                float* __restrict__ q_out)      // [B, VOX, C_B]
{
    const int wave = (blockIdx.x * blockDim.x + threadIdx.x) >> 5;
    const int lane = threadIdx.x & 31;
    const int tilesPerBatch = VOX / 16;          // 864
    const int bat  = wave / tilesPerBatch;
    const int vbase = (wave % tilesPerBatch) * 16;

    const int m16 = lane & 15;
    const int hi  = lane >> 4;                   // 0: K=k0,k0+1  1: K=k0+2,k0+3

    v8f acc0 = {};
    v8f acc1 = {};
    const float* ctb = ct + (size_t)bat * C_CT * VOX;

    for (int k0 = 0; k0 < C_CT; k0 += 4) {
        const int klo = k0 + hi * 2;
        v2f a;
        a.x = ctb[(size_t)klo * VOX + vbase + m16];
        a.y = ctb[(size_t)(klo + 1) * VOX + vbase + m16];
        // B fragment: B[k][n] = qw[n*C_CT + k]; klo even -> 8B aligned float2
        v2f b0 = *(const v2f*)(qw + (size_t)m16 * C_CT + klo);
        v2f b1 = *(const v2f*)(qw + (size_t)(16 + m16) * C_CT + klo);
        acc0 = __builtin_amdgcn_wmma_f32_16x16x4_f32(
                   false, a, false, b0, (short)0, acc0, false, false);
        acc1 = __builtin_amdgcn_wmma_f32_16x16x4_f32(
                   false, a, false, b1, (short)0, acc1, false, false);
    }

    const float bias0 = qb[m16];
    const float bias1 = qb[16 + m16];
    float* qo = q_out + ((size_t)bat * VOX + vbase) * C_B;
#pragma unroll
    for (int r = 0; r < 8; ++r) {
        const int m = r + hi * 8;                // C/D layout: lanes16-31 hold M+8
        qo[(size_t)m * C_B + m16]      = acc0[r] + bias0;
        qo[(size_t)m * C_B + 16 + m16] = acc1[r] + bias1;
    }
}

// ---------------------------------------------------------------------------
// Kernel 2: single-pass online softmax over an N-segment.
// One lane = one voxel; consecutive lanes = consecutive v -> all beam loads
// coalesced. beam is a 637MB one-shot stream -> non-temporal loads so it does
// not thrash the 192MB L2 (logits + partials must stay resident for K3/K4).
// Emits raw logits (to d_out attn region) + partial (m, s, acc[32]).
// ---------------------------------------------------------------------------
__global__ __launch_bounds__(256)
void attn_partial(const float* __restrict__ beam,   // [B, N, C_B, VOX]
                  const float* __restrict__ q,      // [B, VOX, C_B]
                  float* __restrict__ logits_out,   // [B, VOX, N]
                  float* __restrict__ pacc,         // [B*SEG, C_B, VOX]
                  float* __restrict__ pm,           // [B*SEG, VOX]
                  float* __restrict__ ps)           // [B*SEG, VOX]
{
    const int gid = blockIdx.x * blockDim.x + threadIdx.x;
    const int v   = gid % VOX;
    const int bs  = gid / VOX;                   // b*SEG + s
    const int s   = bs % SEG;
    const int bat = bs / SEG;
    const int n0  = s * NSEG;

    // per-lane contiguous 128B of q -> 8 x float4
    float qr[C_B];
    {
        const float4* qp = (const float4*)(q + ((size_t)bat * VOX + v) * C_B);
#pragma unroll
        for (int i = 0; i < 8; ++i) {
            float4 t = qp[i];
            qr[4*i+0] = t.x; qr[4*i+1] = t.y; qr[4*i+2] = t.z; qr[4*i+3] = t.w;
        }
    }

    const float scale = 0.1767766952966369f;     // 32^-0.5
    float mrun = -INFINITY;
    float srun = 0.0f;
    float acc[C_B];
#pragma unroll
    for (int c = 0; c < C_B; ++c) acc[c] = 0.0f;

    float* lptr = logits_out + ((size_t)bat * VOX + v) * NCTRL + n0;
    const float* kbase = beam + ((size_t)(bat * NCTRL + n0) * C_B) * VOX + v;

    for (int n = 0; n < NSEG; ++n) {
        const float* kp = kbase + (size_t)n * C_B * VOX;
        float kv[C_B];
#pragma unroll
        for (int c = 0; c < C_B; ++c)
            kv[c] = __builtin_nontemporal_load(kp + (size_t)c * VOX);
        float dot = 0.0f;
#pragma unroll
        for (int c = 0; c < C_B; ++c) dot = fmaf(qr[c], kv[c], dot);
        const float logit = dot * scale;
        lptr[n] = logit;                          // RT policy: stays in L2 for K4

        const float mnew = fmaxf(mrun, logit);
        const float corr = __expf(mrun - mnew);   // 0 on first iter (exp(-inf))
        const float p    = __expf(logit - mnew);
        srun = fmaf(srun, corr, p);
#pragma unroll
        for (int c = 0; c < C_B; ++c) acc[c] = fmaf(acc[c], corr, p * kv[c]);
        mrun = mnew;
    }

    const size_t idx = (size_t)bs * VOX + v;
    pm[idx] = mrun;
    ps[idx] = srun;
    float* pa = pacc + (size_t)bs * C_B * VOX + v;
#pragma unroll
    for (int c = 0; c < C_B; ++c) pa[(size_t)c * VOX] = acc[c];
}

// ---------------------------------------------------------------------------
// Kernel 3: combine the SEG partials, divide by total sum, LayerNorm over C_B
// (all per-lane: 32 channels live in registers), write fused output and the
// per-voxel (m, 1/s) needed to normalize the stored logits.
// ---------------------------------------------------------------------------
__global__ __launch_bounds__(256)
void attn_combine(const float* __restrict__ pacc,
                  const float* __restrict__ pm,
                  const float* __restrict__ ps,
                  const float* __restrict__ gamma,
                  const float* __restrict__ beta,
                  float* __restrict__ fused_out,  // [B, C_B, VOX]
                  float* __restrict__ mfin,       // [B*VOX]
                  float* __restrict__ isfin)      // [B*VOX]
{
    const int gid = blockIdx.x * blockDim.x + threadIdx.x;  // b*VOX + v
    const int v   = gid % VOX;
    const int bat = gid / VOX;

    float pmv[SEG];
    float m = -INFINITY;
#pragma unroll
    for (int s = 0; s < SEG; ++s) {
        pmv[s] = pm[(size_t)(bat * SEG + s) * VOX + v];
        m = fmaxf(m, pmv[s]);
    }

    float stot = 0.0f;
    float acc[C_B];
#pragma unroll
    for (int c = 0; c < C_B; ++c) acc[c] = 0.0f;
#pragma unroll
    for (int s = 0; s < SEG; ++s) {
        const float w = __expf(pmv[s] - m);
        stot = fmaf(ps[(size_t)(bat * SEG + s) * VOX + v], w, stot);
        const float* pa = pacc + (size_t)(bat * SEG + s) * C_B * VOX + v;
#pragma unroll
        for (int c = 0; c < C_B; ++c) acc[c] = fmaf(pa[(size_t)c * VOX], w, acc[c]);
    }

    const float inv_s = 1.0f / stot;
    float fused[C_B];
    float mu = 0.0f;
#pragma unroll
    for (int c = 0; c < C_B; ++c) { fused[c] = acc[c] * inv_s; mu += fused[c]; }
    mu *= (1.0f / C_B);
    float var = 0.0f;
#pragma unroll
    for (int c = 0; c < C_B; ++c) { const float d = fused[c] - mu; var = fmaf(d, d, var); }
    var *= (1.0f / C_B);
    const float rstd = rsqrtf(var + 1e-5f);

    float* fo = fused_out + (size_t)bat * C_B * VOX + v;
#pragma unroll
    for (int c = 0; c < C_B; ++c)
        fo[(size_t)c * VOX] = (fused[c] - mu) * rstd * gamma[c] + beta[c];

    mfin[gid]  = m;
    isfin[gid] = inv_s;
}

// ---------------------------------------------------------------------------
// Kernel 4: normalize stored logits in place -> attention weights.
// Logits + (m, 1/s) are L2-resident (written by K2/K3, ~21 MB total).
// ---------------------------------------------------------------------------
__global__ __launch_bounds__(256)
void attn_norm(float* __restrict__ attn,          // [B, VOX, N] logits->weights
               const float* __restrict__ mfin,
               const float* __restrict__ isfin)
{
    const size_t t = (size_t)blockIdx.x * blockDim.x + threadIdx.x;
    const size_t total = (size_t)BATCH * VOX * NCTRL;
    if (t >= total) return;
    const int bv = (int)(t / NCTRL);
    attn[t] = __expf(attn[t] - mfin[bv]) * isfin[bv];
}

// ---------------------------------------------------------------------------
extern "C" void kernel_launch(void* const* d_in, const int* in_sizes, int n_in,
                              void* d_out, int out_size, void* d_ws, size_t ws_size,
                              hipStream_t stream) {
    const float* ct    = (const float*)d_in[0];  // [B,128,V]
    const float* beam  = (const float*)d_in[1];  // [B,180,32,V]
    const float* qw    = (const float*)d_in[2];  // [32,128]
    const float* qb    = (const float*)d_in[3];  // [32]
    const float* gamma = (const float*)d_in[4];  // [32]
    const float* beta  = (const float*)d_in[5];  // [32]

    float* out_fused = (float*)d_out;                                   // B*32*V
    float* out_attn  = out_fused + (size_t)BATCH * C_B * VOX;           // B*V*180

    // workspace layout (floats); total ~40 MB
    float* ws   = (float*)d_ws;
    float* q_ws  = ws;                                                  //   884736
    float* pacc  = q_ws  + (size_t)BATCH * VOX * C_B;                   //  8847360
    float* pm    = pacc  + (size_t)BATCH * SEG * C_B * VOX;             //   276480
    float* ps    = pm    + (size_t)BATCH * SEG * VOX;                   //   276480
    float* mfin  = ps    + (size_t)BATCH * SEG * VOX;                   //    27648
    float* isfin = mfin  + (size_t)BATCH * VOX;                         //    27648

    // K1: WMMA query projection. waves = B * V/16 = 1728 -> 216 blocks x 256
    qproj_wmma<<<(BATCH * (VOX / 16) * 32) / 256, 256, 0, stream>>>(ct, qw, qb, q_ws);

    // K2: online-softmax streaming pass. lanes = B*SEG*V -> 1080 blocks
    attn_partial<<<(BATCH * SEG * VOX) / 256, 256, 0, stream>>>(
        beam, q_ws, out_attn, pacc, pm, ps);

    // K3: combine partials + LayerNorm. lanes = B*V -> 108 blocks
    attn_combine<<<(BATCH * VOX) / 256, 256, 0, stream>>>(
        pacc, pm, ps, gamma, beta, out_fused, mfin, isfin);

    // K4: logits -> softmax weights in place. B*V*N = 4976640 -> 19440 blocks
    attn_norm<<<(BATCH * VOX * NCTRL + 255) / 256, 256, 0, stream>>>(
        out_attn, mfin, isfin);
}